// IntraClassCorrelation_82351702933621
// MI455X (gfx1250) — compile-verified
//
#include <hip/hip_runtime.h>
#include <hip/hip_bf16.h>
#include <math.h>

typedef __attribute__((ext_vector_type(2))) float v2f;
typedef __attribute__((ext_vector_type(8))) float v8f;

#define TILE_M 16
#define FEAT_D 128

// ---------------------------------------------------------------------------
// Stage 1: one wave32 per 16-row tile. Per lane: row = lane&15,
// K-pair offset = 4*s + 2*(lane>>4)  -> matches V_WMMA_F32_16X16X4_F32
// A-operand layout (VGPR0=K0/K2, VGPR1=K1/K3). B = all-ones so
// D[m,n] = sum_k diff2[m,k] for every n: WMMA acts as a 16-row horizontal
// reducer in full f32. Accumulate 32 K-slices (D=128).
// ---------------------------------------------------------------------------
__global__ void icc_stage1_dist(const float* __restrict__ anchors,
                                const float* __restrict__ X,
                                const int*   __restrict__ seg,
                                float*       __restrict__ dist,
                                int total)
{
    const int lane   = threadIdx.x & 31;
    const int m      = lane & 15;       // row within tile
    const int half   = lane >> 4;       // 0 -> K={0,1}, 1 -> K={2,3}
    const int wib    = threadIdx.x >> 5;
    const int wpb    = blockDim.x >> 5;
    const int wave   = blockIdx.x * wpb + wib;
    const int nwaves = gridDim.x * wpb;
    const int ntiles = total / TILE_M;

    const float eps = 1e-4f / (float)FEAT_D;

    v2f ones;
    ones[0] = 1.0f; ones[1] = 1.0f;

    for (int tile = wave; tile < ntiles; tile += nwaves) {
        const int    row = tile * TILE_M + m;
        const int    aid = seg[row];
        const float* xr  = X       + (size_t)row * FEAT_D;
        const float* ar  = anchors + (size_t)aid * FEAT_D;

        // Prefetch the next tile's X rows (global_prefetch_b8): keep the
        // 23.3 TB/s HBM stream ahead of the WMMA loop.
        if (tile + nwaves < ntiles) {
            const float* nx = X + ((size_t)(tile + nwaves) * TILE_M + m) * FEAT_D;
            __builtin_prefetch(nx, 0, 1);
            __builtin_prefetch(nx + 64, 0, 1);
        }

        v8f acc0 = {};
        v8f acc1 = {};

#pragma unroll
        for (int s = 0; s < FEAT_D / 4; s += 2) {
            const int k0 = 4 * s + 2 * half;        // even slice
            const int k1 = 4 * (s + 1) + 2 * half;  // odd slice

            v2f xv0 = *(const v2f*)(xr + k0);
            v2f av0 = *(const v2f*)(ar + k0);
            v2f xv1 = *(const v2f*)(xr + k1);
            v2f av1 = *(const v2f*)(ar + k1);

            v2f d0 = xv0 - av0;
            v2f d1 = xv1 - av1;
            v2f a0 = d0 * d0;
            v2f a1 = d1 * d1;

            acc0 = __builtin_amdgcn_wmma_f32_16x16x4_f32(
                       false, a0, false, ones, (short)0, acc0, false, false);
            acc1 = __builtin_amdgcn_wmma_f32_16x16x4_f32(
                       false, a1, false, ones, (short)0, acc1, false, false);
        }

        // D layout: VGPR r -> M=r (lanes 0-15) / M=r+8 (lanes 16-31);
        // every column N holds the same row-sum (B was all ones).
        if ((lane & 15) == 0) {
            const int rbase = tile * TILE_M + half * 8;
#pragma unroll
            for (int r = 0; r < 8; ++r) {
                float sq = acc0[r] + acc1[r];
                dist[rbase + r] = sqrtf(sq + eps);
            }
        }
    }
}

// ---------------------------------------------------------------------------
// Stage 2: one block per anchor. Binary-search the sorted segment_ids for
// this anchor's [start,end), fixed-order strided sum + LDS tree reduction,
// write log1p(per_anchor_sum). Deterministic (no float atomics).
// ---------------------------------------------------------------------------
__device__ __forceinline__ int lower_bound_i32(const int* __restrict__ a,
                                               int n, int key)
{
    int lo = 0, hi = n;
    while (lo < hi) {
        int mid = (lo + hi) >> 1;
        if (a[mid] < key) lo = mid + 1; else hi = mid;
    }
    return lo;
}

__global__ void icc_stage2_segsum(const int*   __restrict__ seg,
                                  const float* __restrict__ dist,
                                  float*       __restrict__ logp,
                                  int total)
{
    const int a     = blockIdx.x;
    const int start = lower_bound_i32(seg, total, a);
    const int end   = lower_bound_i32(seg, total, a + 1);

    float s = 0.0f;
    for (int i = start + threadIdx.x; i < end; i += blockDim.x)
        s += dist[i];

    __shared__ float sm[256];
    sm[threadIdx.x] = s;
    __syncthreads();
#pragma unroll
    for (int off = 128; off > 0; off >>= 1) {
        if ((int)threadIdx.x < off) sm[threadIdx.x] += sm[threadIdx.x + off];
        __syncthreads();
    }
    if (threadIdx.x == 0) logp[a] = log1pf(sm[0]);
}

// ---------------------------------------------------------------------------
// Stage 3: single block: sum the 1024 log1p values, scale by 1/N.
// ---------------------------------------------------------------------------
__global__ void icc_stage3_final(const float* __restrict__ logp,
                                 float*       __restrict__ out,
                                 int num_anchors, float inv_total)
{
    float s = 0.0f;
    for (int i = threadIdx.x; i < num_anchors; i += blockDim.x)
        s += logp[i];

    __shared__ float sm[256];
    sm[threadIdx.x] = s;
    __syncthreads();
#pragma unroll
    for (int off = 128; off > 0; off >>= 1) {
        if ((int)threadIdx.x < off) sm[threadIdx.x] += sm[threadIdx.x + off];
        __syncthreads();
    }
    if (threadIdx.x == 0) out[0] = sm[0] * inv_total;
}

// ---------------------------------------------------------------------------
extern "C" void kernel_launch(void* const* d_in, const int* in_sizes, int n_in,
                              void* d_out, int out_size, void* d_ws, size_t ws_size,
                              hipStream_t stream)
{
    const float* anchors = (const float*)d_in[0];   // [A, D]
    const float* X       = (const float*)d_in[1];   // [TOTAL, D]
    const int*   seg     = (const int*)  d_in[2];   // [TOTAL]
    float*       out     = (float*)d_out;

    const int total = in_sizes[2];                  // 524288
    const int feat  = in_sizes[1] / total;          // 128
    const int nanch = in_sizes[0] / feat;           // 1024
    (void)feat; (void)n_in; (void)out_size; (void)ws_size;

    float* dist = (float*)d_ws;                     // [total]
    float* logp = dist + total;                     // [nanch]

    // Stage 1: 2048 blocks x 256 threads = 16384 waves; 32768 tiles -> 2/wave.
    dim3 b1(256), g1(2048);
    icc_stage1_dist<<<g1, b1, 0, stream>>>(anchors, X, seg, dist, total);

    // Stage 2: one block per anchor.
    dim3 b2(256), g2(nanch);
    icc_stage2_segsum<<<g2, b2, 0, stream>>>(seg, dist, logp, total);

    // Stage 3: single-block final reduction.
    dim3 b3(256), g3(1);
    icc_stage3_final<<<g3, b3, 0, stream>>>(logp, out, nanch,
                                            1.0f / (float)total);
}